// GraphConvSparse_29257317220678
// MI455X (gfx1250) — compile-verified
//
#include <hip/hip_runtime.h>

typedef __attribute__((ext_vector_type(2))) float v2f;
typedef __attribute__((ext_vector_type(8))) float v8f;

#define N_NODES 50000
#define IN_DIM  256
#define OUT_DIM 128
#define N_EDGES 1600000
#define NEG_SLOPE 0.01f

// ---------------------------------------------------------------------------
// Kernel 1: h = X @ W  via V_WMMA_F32_16X16X4_F32 (full f32 precision).
// One wave32 per 16x16 output tile. Block = 256 threads = 8 waves covering
// the 8 N-tiles (8*16 = 128 = OUT_DIM). grid.x = 50000/16 = 3125 M-tiles.
// Two accumulators break the WMMA D->C dependency chain for ILP.
// ---------------------------------------------------------------------------
__global__ __launch_bounds__(256) void gemm_xw_wmma(const float* __restrict__ X,
                                                    const float* __restrict__ W,
                                                    float* __restrict__ H) {
  const int lane = threadIdx.x & 31;
  const int wave = threadIdx.x >> 5;      // N-tile index 0..7
  const int m0   = blockIdx.x * 16;       // 3125 * 16 == 50000 exactly
  const int n0   = wave * 16;
  const int half = lane >> 4;             // lane group 0 / 1
  const int idx  = lane & 15;

  const int m = m0 + idx;                 // A-matrix row for this lane
  const int n = n0 + idx;                 // B-matrix col for this lane

  const float* __restrict__ xrow = X + (size_t)m * IN_DIM;

  v8f acc0 = {};
  v8f acc1 = {};

  for (int k = 0; k < IN_DIM; k += 8) {
    // ---- chunk 0: K = k .. k+3 ----
    {
      const int ka = k + 2 * half;                       // even -> 8B aligned
      v2f a = *(const v2f*)(xrow + ka);                  // A[m][ka], A[m][ka+1]
      v2f b;
      b.x = W[(size_t)ka * OUT_DIM + n];                 // B[ka][n]
      b.y = W[(size_t)(ka + 1) * OUT_DIM + n];           // B[ka+1][n]
      acc0 = __builtin_amdgcn_wmma_f32_16x16x4_f32(
          false, a, false, b, (short)0, acc0, false, false);
    }
    // ---- chunk 1: K = k+4 .. k+7 (independent accumulator) ----
    {
      const int ka = k + 4 + 2 * half;
      v2f a = *(const v2f*)(xrow + ka);
      v2f b;
      b.x = W[(size_t)ka * OUT_DIM + n];
      b.y = W[(size_t)(ka + 1) * OUT_DIM + n];
      acc1 = __builtin_amdgcn_wmma_f32_16x16x4_f32(
          false, a, false, b, (short)0, acc1, false, false);
    }
  }

  // C/D layout: VGPR v -> row m0 + v + 8*half, col n0 + idx
#pragma unroll
  for (int v = 0; v < 8; ++v) {
    const int row = m0 + v + 8 * half;
    H[(size_t)row * OUT_DIM + n] = acc0[v] + acc1[v];
  }
}

// ---------------------------------------------------------------------------
// Kernel 2: zero the aggregation buffer (float4-wide).
// ---------------------------------------------------------------------------
__global__ __launch_bounds__(256) void zero_f4(float4* __restrict__ p, int n4) {
  int i = blockIdx.x * blockDim.x + threadIdx.x;
  if (i < n4) p[i] = make_float4(0.f, 0.f, 0.f, 0.f);
}

// ---------------------------------------------------------------------------
// Kernel 3: edge scatter. One wave32 per edge. Edge metadata (r, c, v) is
// wave-uniform: force it through readfirstlane so the compiler emits scalar
// s_load_b32 (constant cache) instead of 32-lane broadcast vector loads.
// Lanes then cooperatively read the 128-float row h[c] (coalesced b128,
// L2-resident) and scatter v * h[c] into out[r] with native
// global_atomic_add_f32 (unsafeAtomicAdd).
// ---------------------------------------------------------------------------
__global__ __launch_bounds__(256) void spmm_scatter(const float* __restrict__ H,
                                                    const float* __restrict__ vals,
                                                    const int* __restrict__ rows,
                                                    const int* __restrict__ cols,
                                                    float* __restrict__ out) {
  const int lane = threadIdx.x & 31;
  // wave-uniform edge index, made provably uniform for SGPR promotion
  const int e = __builtin_amdgcn_readfirstlane(blockIdx.x * 8 + (threadIdx.x >> 5));
  if (e >= N_EDGES) return;

  const int   r = rows[e];   // -> s_load_b32
  const int   c = cols[e];   // -> s_load_b32
  const float v = vals[e];   // -> s_load_b32

  const float4 hv = ((const float4*)(H + (size_t)c * OUT_DIM))[lane];
  float* __restrict__ dst = out + (size_t)r * OUT_DIM + lane * 4;

  unsafeAtomicAdd(dst + 0, v * hv.x);
  unsafeAtomicAdd(dst + 1, v * hv.y);
  unsafeAtomicAdd(dst + 2, v * hv.z);
  unsafeAtomicAdd(dst + 3, v * hv.w);
}

// ---------------------------------------------------------------------------
// Kernel 4: leaky-ReLU epilogue in place (float4-wide).
// ---------------------------------------------------------------------------
__global__ __launch_bounds__(256) void leaky_relu_f4(float4* __restrict__ p, int n4) {
  int i = blockIdx.x * blockDim.x + threadIdx.x;
  if (i < n4) {
    float4 v = p[i];
    v.x = v.x >= 0.f ? v.x : NEG_SLOPE * v.x;
    v.y = v.y >= 0.f ? v.y : NEG_SLOPE * v.y;
    v.z = v.z >= 0.f ? v.z : NEG_SLOPE * v.z;
    v.w = v.w >= 0.f ? v.w : NEG_SLOPE * v.w;
    p[i] = v;
  }
}

// ---------------------------------------------------------------------------
// Host entry. Inputs (setup_inputs order): x, weight, adj_vals, adj_rows,
// adj_cols. Output: [N_NODES, OUT_DIM] f32. Workspace holds h (25.6 MB).
// ---------------------------------------------------------------------------
extern "C" void kernel_launch(void* const* d_in, const int* in_sizes, int n_in,
                              void* d_out, int out_size, void* d_ws, size_t ws_size,
                              hipStream_t stream) {
  const float* x    = (const float*)d_in[0];
  const float* w    = (const float*)d_in[1];
  const float* vals = (const float*)d_in[2];
  const int*   rows = (const int*)d_in[3];
  const int*   cols = (const int*)d_in[4];
  float*       out  = (float*)d_out;
  float*       h    = (float*)d_ws;   // 50000*128 floats = 25.6 MB scratch

  const int n_out   = N_NODES * OUT_DIM;       // 6,400,000
  const int n_out4  = n_out / 4;               // 1,600,000
  const int zb      = (n_out4 + 255) / 256;    // 6250 blocks

  zero_f4<<<zb, 256, 0, stream>>>((float4*)out, n_out4);

  gemm_xw_wmma<<<N_NODES / 16, 256, 0, stream>>>(x, w, h);   // 3125 blocks

  spmm_scatter<<<N_EDGES / 8, 256, 0, stream>>>(h, vals, rows, cols, out);

  leaky_relu_f4<<<zb, 256, 0, stream>>>((float4*)out, n_out4);
}